// Step21TwoPathLocalSubgraphChooser_19344532702121
// MI455X (gfx1250) — compile-verified
//
#include <hip/hip_runtime.h>
#include <hip/hip_bf16.h>

typedef __attribute__((ext_vector_type(2))) float v2f;
typedef __attribute__((ext_vector_type(8))) float v8f;

#define B_  8
#define N_  64
#define PD  64
#define ND  128
#define EMB 48
#define HID 96
#define HIN 452   // PD + 8*EMB + 4

__device__ __forceinline__ v8f wmma4(v2f a, v2f b, v8f c) {
    // D = A(16x4 f32) * B(4x16 f32) + C(16x16 f32)
    return __builtin_amdgcn_wmma_f32_16x16x4_f32(
        false, a, false, b, (short)0, c, false, false);
}

// ---------------------------------------------------------------------------
// Kernel A: node_emb = relu(X @ nW0 + nb0) @ nW1 + nb1     (per batch)
// ---------------------------------------------------------------------------
__global__ __launch_bounds__(256) void node_emb_kernel(
    const float* __restrict__ X_,     // B,64,128
    const float* __restrict__ nW0,    // 128,96
    const float* __restrict__ nb0,    // 96
    const float* __restrict__ nW1,    // 96,48
    const float* __restrict__ nb1,    // 48
    float* __restrict__ emb)          // B,64,48
{
    __shared__ float h1[N_][HID];     // 24 KB
    const int b    = blockIdx.x;
    const int tid  = threadIdx.x;
    const int wave = tid >> 5;
    const int lane = tid & 31;
    const int row  = lane & 15;
    const int khalf = (lane >> 4) << 1;   // 0 or 2
    const int mhalf = (lane >> 4) << 3;   // 0 or 8
    const float* X = X_ + b * N_ * ND;

    // Layer 0: 4 M-tiles x 6 N-tiles = 24 tiles over 8 waves
    for (int t = wave; t < 24; t += 8) {
        const int m0 = (t & 3) * 16;
        const int n0 = (t >> 2) * 16;
        v8f acc = {};
        for (int k = 0; k < ND; k += 4) {
            const int kk = k + khalf;
            v2f a, bm;
            a.x  = X[(m0 + row) * ND + kk];
            a.y  = X[(m0 + row) * ND + kk + 1];
            bm.x = nW0[kk * HID + n0 + row];
            bm.y = nW0[(kk + 1) * HID + n0 + row];
            acc = wmma4(a, bm, acc);
        }
        for (int r = 0; r < 8; ++r) {
            const int mr = m0 + r + mhalf;
            const int nc = n0 + row;
            float v = acc[r] + nb0[nc];
            h1[mr][nc] = v > 0.f ? v : 0.f;
        }
    }
    __syncthreads();

    // Layer 1: 4 M-tiles x 3 N-tiles = 12 tiles over 8 waves
    for (int t = wave; t < 12; t += 8) {
        const int m0 = (t & 3) * 16;
        const int n0 = (t >> 2) * 16;
        v8f acc = {};
        for (int k = 0; k < HID; k += 4) {
            const int kk = k + khalf;
            v2f a, bm;
            a.x  = h1[m0 + row][kk];
            a.y  = h1[m0 + row][kk + 1];
            bm.x = nW1[kk * EMB + n0 + row];
            bm.y = nW1[(kk + 1) * EMB + n0 + row];
            acc = wmma4(a, bm, acc);
        }
        for (int r = 0; r < 8; ++r) {
            const int mr = m0 + r + mhalf;
            const int nc = n0 + row;
            emb[(b * N_ + mr) * EMB + nc] = acc[r] + nb1[nc];
        }
    }
}

// ---------------------------------------------------------------------------
// Kernel B: deg[b,i] = sum_k adj[b,i,k]*scoped[k] / clip(sum scoped,1)
// ---------------------------------------------------------------------------
__global__ __launch_bounds__(64) void deg_kernel(
    const float* __restrict__ adj,    // B,64,64
    const int*   __restrict__ scoped, // B,64
    float* __restrict__ deg)          // B,64
{
    __shared__ float sc[N_];
    __shared__ float scount;
    const int b = blockIdx.x;
    const int i = threadIdx.x;
    sc[i] = scoped[b * N_ + i] ? 1.f : 0.f;
    __syncthreads();
    if (i == 0) {
        float s = 0.f;
        for (int k = 0; k < N_; ++k) s += sc[k];
        scount = s < 1.f ? 1.f : s;
    }
    __syncthreads();
    float s = 0.f;
    for (int k = 0; k < N_; ++k) s += adj[(b * N_ + i) * N_ + k] * sc[k];
    deg[b * N_ + i] = s / scount;
}

// ---------------------------------------------------------------------------
// Kernel C: per (b, i, j-tile of 16) build head_in and run the head MLP.
// ---------------------------------------------------------------------------
__global__ __launch_bounds__(256) void head_kernel(
    const float* __restrict__ pair,   // B,64,64,64
    const float* __restrict__ adj,    // B,64,64
    const int*   __restrict__ scoped, // B,64
    const float* __restrict__ emb,    // B,64,48
    const float* __restrict__ deg,    // B,64
    const float* __restrict__ hW0, const float* __restrict__ hb0,
    const float* __restrict__ hW1, const float* __restrict__ hb1,
    const float* __restrict__ hW2, const float* __restrict__ hb2,
    float* __restrict__ rawL)         // B,64,64
{
    __shared__ float emb_s[N_][EMB];          // 12 KB
    __shared__ float adjI[N_];
    __shared__ float sc_s[N_];
    __shared__ float scount;
    __shared__ unsigned char lmask[16][N_];   // 1 KB
    __shared__ unsigned char cmask[16][N_];   // 1 KB
    __shared__ float lmean[16][EMB];
    __shared__ float lmaxv[16][EMB];
    __shared__ float cmean[16][EMB];
    __shared__ float escal[16][8];            // degi,degj,lfrac,cfrac,lcnt,ccnt,lraw
    __shared__ float hin[16][456];            // 29 KB (stride padded)
    __shared__ float h1[16][HID];
    __shared__ float h2[16][HID];

    const int blk = blockIdx.x;               // B*N*4 blocks
    const int b   = blk >> 8;                 // / (64*4)
    const int rem = blk & 255;
    const int i   = rem >> 2;
    const int jt  = rem & 3;
    const int j0  = jt * 16;
    const int tid  = threadIdx.x;
    const int wave = tid >> 5;
    const int lane = tid & 31;
    const int row  = lane & 15;
    const int khalf = (lane >> 4) << 1;
    const int mhalf = (lane >> 4) << 3;

    // ---- load batch-local data ----
    for (int t = tid; t < N_ * EMB; t += 256)
        emb_s[t / EMB][t % EMB] = emb[(b * N_ + t / EMB) * EMB + t % EMB];
    if (tid < N_) {
        adjI[tid] = adj[(b * N_ + i) * N_ + tid];
        sc_s[tid] = scoped[b * N_ + tid] ? 1.f : 0.f;
    }
    __syncthreads();

    // ---- masks (+ scope count on thread 0) ----
    if (tid == 0) {
        float s = 0.f;
        for (int k = 0; k < N_; ++k) s += sc_s[k];
        scount = s < 1.f ? 1.f : s;
    }
    for (int t = tid; t < 16 * N_; t += 256) {
        const int e = t >> 6, k = t & 63;
        const int j = j0 + e;
        const bool ai = adjI[k] > 0.5f;
        const bool aj = adj[(b * N_ + j) * N_ + k] > 0.5f;
        const bool sk = sc_s[k] > 0.5f;
        lmask[e][k] = (sk && (k == i || k == j || ai || aj)) ? 1 : 0;
        cmask[e][k] = (sk && ai && aj) ? 1 : 0;
    }
    __syncthreads();

    // ---- per-edge counts & scalars ----
    if (tid < 16) {
        const int e = tid, j = j0 + e;
        float ls = 0.f, cs = 0.f;
        for (int k = 0; k < N_; ++k) { ls += lmask[e][k]; cs += cmask[e][k]; }
        const float lcnt = ls < 1.f ? 1.f : ls;
        const float ccnt = cs < 1.f ? 1.f : cs;
        escal[e][0] = deg[b * N_ + i];
        escal[e][1] = deg[b * N_ + j];
        escal[e][2] = lcnt / scount;
        escal[e][3] = ccnt / scount;
        escal[e][4] = lcnt;
        escal[e][5] = ccnt;
        escal[e][6] = ls;    // raw local count (for empty-max check)
    }
    __syncthreads();

    // ---- pooled mean / max / common mean over k ----
    for (int t = tid; t < 16 * EMB; t += 256) {
        const int e = t / EMB, d = t % EMB;
        float ls = 0.f, cs = 0.f, mx = -10000.f;
        for (int k = 0; k < N_; ++k) {
            const float v = emb_s[k][d];
            if (lmask[e][k]) { ls += v; if (v > mx) mx = v; }
            if (cmask[e][k]) cs += v;
        }
        lmean[e][d] = ls / escal[e][4];
        cmean[e][d] = cs / escal[e][5];
        lmaxv[e][d] = (escal[e][6] <= 0.f) ? 0.f : mx;
    }
    __syncthreads();

    // ---- assemble head_in (16 x 452) ----
    for (int t = tid; t < 16 * HIN; t += 256) {
        const int e = t / HIN, c = t % HIN;
        const int j = j0 + e;
        float v;
        if (c < 64) {
            v = pair[(((b * N_ + i) * N_ + j) * PD) + c];
        } else if (c < 112) {
            v = emb_s[i][c - 64];
        } else if (c < 160) {
            v = emb_s[j][c - 112];
        } else if (c < 208) {
            v = fabsf(emb_s[i][c - 160] - emb_s[j][c - 160]);
        } else if (c < 256) {
            v = emb_s[i][c - 208] * emb_s[j][c - 208];
        } else if (c < 304) {
            v = lmean[e][c - 256];
        } else if (c < 352) {
            v = lmaxv[e][c - 304];
        } else if (c < 400) {
            v = cmean[e][c - 352];
        } else if (c < 448) {
            v = fabsf(lmean[e][c - 400] - cmean[e][c - 400]);
        } else {
            v = escal[e][c - 448];
        }
        hin[e][c] = v;
    }
    __syncthreads();

    // ---- layer 0: (16 x 452) @ (452 x 96), 6 N-tiles across waves 0..5 ----
    if (wave < 6) {
        const int n0 = wave * 16;
        v8f acc = {};
        for (int k = 0; k < HIN; k += 4) {
            const int kk = k + khalf;
            v2f a, bm;
            a.x  = hin[row][kk];
            a.y  = hin[row][kk + 1];
            bm.x = hW0[kk * HID + n0 + row];
            bm.y = hW0[(kk + 1) * HID + n0 + row];
            acc = wmma4(a, bm, acc);
        }
        for (int r = 0; r < 8; ++r) {
            const int mr = r + mhalf;
            const int nc = n0 + row;
            float v = acc[r] + hb0[nc];
            h1[mr][nc] = v > 0.f ? v : 0.f;
        }
    }
    __syncthreads();

    // ---- layer 1: (16 x 96) @ (96 x 96) ----
    if (wave < 6) {
        const int n0 = wave * 16;
        v8f acc = {};
        for (int k = 0; k < HID; k += 4) {
            const int kk = k + khalf;
            v2f a, bm;
            a.x  = h1[row][kk];
            a.y  = h1[row][kk + 1];
            bm.x = hW1[kk * HID + n0 + row];
            bm.y = hW1[(kk + 1) * HID + n0 + row];
            acc = wmma4(a, bm, acc);
        }
        for (int r = 0; r < 8; ++r) {
            const int mr = r + mhalf;
            const int nc = n0 + row;
            float v = acc[r] + hb1[nc];
            h2[mr][nc] = v > 0.f ? v : 0.f;
        }
    }
    __syncthreads();

    // ---- layer 2: (16 x 96) @ (96 x 1) ----
    if (tid < 16) {
        const int e = tid, j = j0 + e;
        float s = 0.f;
        for (int k = 0; k < HID; ++k) s += h2[e][k] * hW2[k];
        rawL[(b * N_ + i) * N_ + j] = s + hb2[0];
    }
}

// ---------------------------------------------------------------------------
// Kernel D: symmetrize + mask
// ---------------------------------------------------------------------------
__global__ __launch_bounds__(256) void sym_kernel(
    const float* __restrict__ rawL,
    const unsigned char* __restrict__ valid,
    float* __restrict__ out)
{
    const int idx = blockIdx.x * 256 + threadIdx.x;
    if (idx >= B_ * N_ * N_) return;
    const int b = idx >> 12;
    const int r = idx & 4095;
    const int i = r >> 6;
    const int j = r & 63;
    const float v = 0.5f * (rawL[idx] + rawL[(b * N_ + j) * N_ + i]);
    out[idx] = valid[idx] ? v : -1000000000.0f;
}

// ---------------------------------------------------------------------------
extern "C" void kernel_launch(void* const* d_in, const int* in_sizes, int n_in,
                              void* d_out, int out_size, void* d_ws, size_t ws_size,
                              hipStream_t stream) {
    const float* pair   = (const float*)d_in[0];
    const float* nodef  = (const float*)d_in[1];
    const float* adj    = (const float*)d_in[2];
    const int*   scoped = (const int*)  d_in[3];
    const unsigned char* valid = (const unsigned char*)d_in[4];
    const float* nW0 = (const float*)d_in[5];
    const float* nb0 = (const float*)d_in[6];
    const float* nW1 = (const float*)d_in[7];
    const float* nb1 = (const float*)d_in[8];
    const float* hW0 = (const float*)d_in[9];
    const float* hb0 = (const float*)d_in[10];
    const float* hW1 = (const float*)d_in[11];
    const float* hb1 = (const float*)d_in[12];
    const float* hW2 = (const float*)d_in[13];
    const float* hb2 = (const float*)d_in[14];

    float* ws   = (float*)d_ws;
    float* emb  = ws;                         // B*64*48   = 24576
    float* deg  = emb + B_ * N_ * EMB;        // B*64      = 512
    float* rawL = deg + B_ * N_;              // B*64*64   = 32768

    node_emb_kernel<<<B_, 256, 0, stream>>>(nodef, nW0, nb0, nW1, nb1, emb);
    deg_kernel<<<B_, 64, 0, stream>>>(adj, scoped, deg);
    head_kernel<<<B_ * N_ * 4, 256, 0, stream>>>(
        pair, adj, scoped, emb, deg,
        hW0, hb0, hW1, hb1, hW2, hb2, rawL);
    sym_kernel<<<(B_ * N_ * N_ + 255) / 256, 256, 0, stream>>>(
        rawL, valid, (float*)d_out);
}